// ScopeSparse_38929583571237
// MI455X (gfx1250) — compile-verified
//
#include <hip/hip_runtime.h>
#include <math.h>

// Problem constants (match reference)
#define BB 32
#define LL 4096
#define CC 64
#define DD 128
#define KK 2048          // TOPK = L/2
#define NT (KK / 16)     // 128 sparse 16-row tiles per batch
#define MT (LL / 16)     // 256 full 16-row tiles per batch

typedef __attribute__((ext_vector_type(2))) float v2f;
typedef __attribute__((ext_vector_type(8))) float v8f;

// gfx1250 async global->LDS path (ASYNCcnt-tracked), guarded so the file
// still compiles on toolchains that do not declare the builtins.
// Probe (round 2 diagnostic) says arg0 is `int __vector(4) __device__ *`,
// i.e. a non-const v4i pointer in AS(1); arg1 analogously AS(3).
#if __has_builtin(__builtin_amdgcn_global_load_async_to_lds_b128)
#define HAVE_ASYNC_LDS 1
typedef int a_v4i __attribute__((vector_size(16)));
typedef __attribute__((address_space(1))) a_v4i* g_v4i_p;
typedef __attribute__((address_space(3))) a_v4i* l_v4i_p;
__device__ __forceinline__ void async_tile_b128(const float* gsrc, float* ldst) {
    __builtin_amdgcn_global_load_async_to_lds_b128(
        (g_v4i_p)(void*)const_cast<float*>(gsrc),
        (l_v4i_p)(void*)ldst, 0, 0);
}
__device__ __forceinline__ void async_wait0() {
#if __has_builtin(__builtin_amdgcn_s_wait_asynccnt)
    __builtin_amdgcn_s_wait_asynccnt(0);
#else
    asm volatile("s_wait_asynccnt 0x0" ::: "memory");
#endif
}
#else
#define HAVE_ASYNC_LDS 0
#endif

// Monotonic float -> uint transform (larger float <=> larger uint)
__device__ __forceinline__ unsigned f2ord(float f) {
    unsigned u = __float_as_uint(f);
    return u ^ ((u >> 31) ? 0xFFFFFFFFu : 0x80000000u);
}

// ---------------------------------------------------------------------------
// Kernel 1: scores[b,l] = dot(x[b,l,:], score_w) + score_b   (one wave / row)
// ---------------------------------------------------------------------------
__global__ void __launch_bounds__(256) k_scores(const float* __restrict__ x,
                                                const float* __restrict__ sw,
                                                const float* __restrict__ sb,
                                                float* __restrict__ scores) {
    int row  = blockIdx.x * 8 + (threadIdx.x >> 5);
    int lane = threadIdx.x & 31;
    if (row >= BB * LL) return;
    float2 xv = *reinterpret_cast<const float2*>(x + (size_t)row * CC + lane * 2);
    float2 wv = *reinterpret_cast<const float2*>(sw + lane * 2);
    float s = xv.x * wv.x + xv.y * wv.y;
#pragma unroll
    for (int m = 16; m >= 1; m >>= 1) s += __shfl_xor(s, m, 32);
    if (lane == 0) scores[row] = s + sb[0];
}

// ---------------------------------------------------------------------------
// Kernel 2: per-batch radix select of K-th largest score (threshold + ties)
// meta[b] = { T (ordered-uint), cntAbove, tieQuota, 0 }
// ---------------------------------------------------------------------------
__global__ void __launch_bounds__(256) k_select(const float* __restrict__ scores,
                                                unsigned* __restrict__ meta) {
    __shared__ unsigned hist[256];
    __shared__ unsigned sPrefix, sK, sCnt;
    const int b   = blockIdx.x;
    const int tid = threadIdx.x;
    const float* sc = scores + (size_t)b * LL;
    if (tid == 0) { sPrefix = 0u; sK = KK; }
    for (int p = 3; p >= 0; --p) {
        hist[tid] = 0u;
        __syncthreads();
        unsigned pref = sPrefix;
        for (int i = tid; i < LL; i += 256) {
            unsigned u = f2ord(sc[i]);
            bool match = (p == 3) || ((u >> ((p + 1) * 8)) == pref);
            if (match) atomicAdd(&hist[(u >> (p * 8)) & 0xFFu], 1u);
        }
        __syncthreads();
        if (tid == 0) {
            unsigned k = sK, cum = 0u, chosen = 0u;
            for (int bin = 255; bin >= 0; --bin) {
                unsigned c = hist[bin];
                if (cum + c >= k) { chosen = (unsigned)bin; sK = k - cum; break; }
                cum += c;
            }
            sPrefix = (pref << 8) | chosen;
        }
        __syncthreads();
    }
    if (tid == 0) sCnt = 0u;
    __syncthreads();
    unsigned T = sPrefix, cnt = 0u;
    for (int i = tid; i < LL; i += 256)
        if (f2ord(sc[i]) > T) cnt++;
    atomicAdd(&sCnt, cnt);
    __syncthreads();
    if (tid == 0) {
        meta[b * 4 + 0] = T;
        meta[b * 4 + 1] = sCnt;
        meta[b * 4 + 2] = (unsigned)KK - sCnt;
        meta[b * 4 + 3] = 0u;
    }
}

// ---------------------------------------------------------------------------
// Kernel 3: deterministic ordered compaction of selected row indices.
// One wave per batch; ballot/popc prefix within wave, sequential chunks.
// ---------------------------------------------------------------------------
__global__ void __launch_bounds__(32) k_compact(const float* __restrict__ scores,
                                                const unsigned* __restrict__ meta,
                                                int* __restrict__ idxl) {
    const int b    = blockIdx.x;
    const int lane = threadIdx.x;
    const unsigned T        = meta[b * 4 + 0];
    const unsigned cntAbove = meta[b * 4 + 1];
    const unsigned tieQuota = meta[b * 4 + 2];
    const float* sc = scores + (size_t)b * LL;
    int* ol = idxl + (size_t)b * KK;
    unsigned gBase = 0u, eBase = 0u;
    const unsigned lt = (lane == 0) ? 0u : (0xFFFFFFFFu >> (32 - lane));
    for (int chunk = 0; chunk < LL / 32; ++chunk) {
        int row = chunk * 32 + lane;
        unsigned u = f2ord(sc[row]);
        bool pg = (u > T);
        bool pe = (u == T);
        unsigned mg = (unsigned)__ballot(pg);
        unsigned me = (unsigned)__ballot(pe);
        if (pg) ol[gBase + __popc(mg & lt)] = row;
        if (pe) {
            unsigned rank = eBase + __popc(me & lt);
            if (rank < tieQuota) ol[cntAbove + rank] = row;
        }
        gBase += __popc(mg);
        eBase += __popc(me);
    }
}

// ---------------------------------------------------------------------------
// Kernel 4: gathered sparse GEMM tile (16 rows x D) via WMMA f32 16x16x4,
// exact GELU, per-tile column sums -> partial[b][tile][d]  (deterministic)
// ---------------------------------------------------------------------------
__global__ void __launch_bounds__(256) k_sparse(const float* __restrict__ x,
                                                const int* __restrict__ idxl,
                                                const float* __restrict__ wsp,
                                                const float* __restrict__ bsp,
                                                float* __restrict__ partial) {
    __shared__ __align__(16) float xs[16 * 68];
    __shared__ int rows[16];
    const int blk  = blockIdx.x;
    const int b    = blk / NT;
    const int tile = blk % NT;
    const int tid  = threadIdx.x;
    const int lane = tid & 31;
    const int wv   = tid >> 5;          // 8 waves, each owns N-slice of 16
    if (tid < 16) rows[tid] = idxl[(size_t)b * KK + tile * 16 + tid];
    __syncthreads();
    {   // gather A tile (16x64) into LDS: 256 lanes x 16B each
        int r = tid >> 4;
        int c = (tid & 15) * 4;
        const float* src = x + ((size_t)b * LL + rows[r]) * CC + c;
#if HAVE_ASYNC_LDS
        async_tile_b128(src, &xs[r * 68 + c]);   // global_load_async_to_lds_b128
        async_wait0();                           // s_wait_asynccnt 0
#else
        float4 v = *reinterpret_cast<const float4*>(src);
        *reinterpret_cast<float4*>(&xs[r * 68 + c]) = v;
#endif
    }
    __syncthreads();
    const int n0    = wv * 16;
    const int n     = lane & 15;
    const int m     = lane & 15;
    const int khalf = (lane >> 4) * 2;   // lanes 0-15: K+{0,1}; 16-31: K+{2,3}
    v2f bfr[16];
#pragma unroll
    for (int kk = 0; kk < 16; ++kk) {
        int kA = kk * 4 + khalf;
        bfr[kk].x = wsp[(size_t)kA * DD + n0 + n];
        bfr[kk].y = wsp[(size_t)(kA + 1) * DD + n0 + n];
    }
    v8f acc = {0.f, 0.f, 0.f, 0.f, 0.f, 0.f, 0.f, 0.f};
#pragma unroll
    for (int kk = 0; kk < 16; ++kk) {
        int kA = kk * 4 + khalf;
        v2f a;
        a.x = xs[m * 68 + kA];
        a.y = xs[m * 68 + kA + 1];
        acc = __builtin_amdgcn_wmma_f32_16x16x4_f32(
            false, a, false, bfr[kk], (short)0, acc, false, false);
    }
    float bias = bsp[n0 + n];
    float s = 0.f;
#pragma unroll
    for (int r = 0; r < 8; ++r) {
        float v = acc[r] + bias;
        s += 0.5f * v * (1.0f + erff(v * 0.70710678118654752f)); // exact GELU
    }
    s += __shfl_xor(s, 16, 32);          // combine rows 0-7 with rows 8-15
    if (lane < 16) partial[((size_t)b * NT + tile) * DD + n0 + n] = s;
}

// ---------------------------------------------------------------------------
// Kernel 5: fixed-order reduction of tile partials -> sparse_agg (mean)
// ---------------------------------------------------------------------------
__global__ void __launch_bounds__(128) k_reduce(const float* __restrict__ partial,
                                                float* __restrict__ sagg) {
    const int b = blockIdx.x;
    const int d = threadIdx.x;
    float s = 0.f;
    for (int t = 0; t < NT; ++t) s += partial[((size_t)b * NT + t) * DD + d];
    sagg[b * DD + d] = s * (1.0f / (float)KK);
}

// ---------------------------------------------------------------------------
// Kernel 6: full GEMM tile via WMMA + bias + sparse_agg + LayerNorm
// ---------------------------------------------------------------------------
__global__ void __launch_bounds__(256) k_full(const float* __restrict__ x,
                                              const float* __restrict__ wf,
                                              const float* __restrict__ fb,
                                              const float* __restrict__ sagg,
                                              const float* __restrict__ lng,
                                              const float* __restrict__ lnb,
                                              float* __restrict__ out) {
    __shared__ __align__(16) float xs[16 * 68];
    __shared__ __align__(16) float ot[16 * 132];
    const int blk = blockIdx.x;
    const int b   = blk / MT;
    const int l0  = (blk % MT) * 16;
    const int tid  = threadIdx.x;
    const int lane = tid & 31;
    const int wv   = tid >> 5;
    {   // stage x tile (16x64) in LDS: 256 lanes x 16B each
        int r = tid >> 4;
        int c = (tid & 15) * 4;
        const float* src = x + ((size_t)b * LL + l0 + r) * CC + c;
#if HAVE_ASYNC_LDS
        async_tile_b128(src, &xs[r * 68 + c]);   // global_load_async_to_lds_b128
        async_wait0();                           // s_wait_asynccnt 0
#else
        float4 v = *reinterpret_cast<const float4*>(src);
        *reinterpret_cast<float4*>(&xs[r * 68 + c]) = v;
#endif
    }
    __syncthreads();
    const int n0    = wv * 16;
    const int n     = lane & 15;
    const int m     = lane & 15;
    const int khalf = (lane >> 4) * 2;
    v2f bfr[16];
#pragma unroll
    for (int kk = 0; kk < 16; ++kk) {
        int kA = kk * 4 + khalf;
        bfr[kk].x = wf[(size_t)kA * DD + n0 + n];
        bfr[kk].y = wf[(size_t)(kA + 1) * DD + n0 + n];
    }
    v8f acc = {0.f, 0.f, 0.f, 0.f, 0.f, 0.f, 0.f, 0.f};
#pragma unroll
    for (int kk = 0; kk < 16; ++kk) {
        int kA = kk * 4 + khalf;
        v2f a;
        a.x = xs[m * 68 + kA];
        a.y = xs[m * 68 + kA + 1];
        acc = __builtin_amdgcn_wmma_f32_16x16x4_f32(
            false, a, false, bfr[kk], (short)0, acc, false, false);
    }
    float addv = fb[n0 + n] + sagg[b * DD + n0 + n];
#pragma unroll
    for (int r = 0; r < 8; ++r) {
        int row = (lane < 16) ? r : (r + 8);
        ot[row * 132 + n0 + n] = acc[r] + addv;
    }
    __syncthreads();
    // LayerNorm: each wave normalizes 2 rows (8 waves x 2 = 16 rows)
#pragma unroll
    for (int rr = 0; rr < 2; ++rr) {
        int row = wv * 2 + rr;
        float4 v = *reinterpret_cast<const float4*>(&ot[row * 132 + lane * 4]);
        float sum = v.x + v.y + v.z + v.w;
        float sq  = v.x * v.x + v.y * v.y + v.z * v.z + v.w * v.w;
#pragma unroll
        for (int msk = 16; msk >= 1; msk >>= 1) {
            sum += __shfl_xor(sum, msk, 32);
            sq  += __shfl_xor(sq,  msk, 32);
        }
        float mean = sum * (1.0f / (float)DD);
        float var  = sq * (1.0f / (float)DD) - mean * mean;
        float rstd = rsqrtf(var + 1e-5f);
        int col = lane * 4;
        float4 gg = *reinterpret_cast<const float4*>(lng + col);
        float4 bb = *reinterpret_cast<const float4*>(lnb + col);
        float4 o;
        o.x = (v.x - mean) * rstd * gg.x + bb.x;
        o.y = (v.y - mean) * rstd * gg.y + bb.y;
        o.z = (v.z - mean) * rstd * gg.z + bb.z;
        o.w = (v.w - mean) * rstd * gg.w + bb.w;
        *reinterpret_cast<float4*>(out + ((size_t)b * LL + l0 + row) * DD + col) = o;
    }
}

// ---------------------------------------------------------------------------
extern "C" void kernel_launch(void* const* d_in, const int* in_sizes, int n_in,
                              void* d_out, int out_size, void* d_ws, size_t ws_size,
                              hipStream_t stream) {
    const float* x        = (const float*)d_in[0];
    const float* score_w  = (const float*)d_in[1];
    const float* score_b  = (const float*)d_in[2];
    const float* sparse_w = (const float*)d_in[3];
    const float* sparse_b = (const float*)d_in[4];
    const float* full_w   = (const float*)d_in[5];
    const float* full_b   = (const float*)d_in[6];
    const float* ln_g     = (const float*)d_in[7];
    const float* ln_b     = (const float*)d_in[8];
    float* out = (float*)d_out;

    // Workspace layout (all 4-byte types): ~2.9 MB total
    float*    scores  = (float*)d_ws;                         // B*L
    unsigned* meta    = (unsigned*)(scores + BB * LL);        // B*4
    int*      idxl    = (int*)(meta + BB * 4);                // B*K
    float*    partial = (float*)(idxl + BB * KK);             // B*NT*D
    float*    sagg    = partial + (size_t)BB * NT * DD;       // B*D

    k_scores <<<BB * LL / 8, 256, 0, stream>>>(x, score_w, score_b, scores);
    k_select <<<BB,          256, 0, stream>>>(scores, meta);
    k_compact<<<BB,           32, 0, stream>>>(scores, meta, idxl);
    k_sparse <<<BB * NT,     256, 0, stream>>>(x, idxl, sparse_w, sparse_b, partial);
    k_reduce <<<BB,          128, 0, stream>>>(partial, sagg);
    k_full   <<<BB * MT,     256, 0, stream>>>(x, full_w, full_b, sagg, ln_g, ln_b, out);
}